// M_layer_4818953306926
// MI455X (gfx1250) — compile-verified
//
#include <hip/hip_runtime.h>

typedef __attribute__((ext_vector_type(2))) float v2f;
typedef __attribute__((ext_vector_type(8))) float v8f;

#define PI_F      3.14159265358979323846f
#define LOG2E_F   1.44269504088896340736f
// base-2 exponent = K1*lam - K2*phi_m - K2*phi_x
#define K1_F (4.0f * PI_F * LOG2E_F)
#define K2_F (2.0f * PI_F * LOG2E_F)

#define NDOM   16384
#define NPTS   1024
#define NBATCH 16
#define SETS   17                       // 16 batches of C1 + 1 set of C
#define KTILES (NDOM / 16)              // 1024
#define TILES_PER_SET (NPTS / 16)       // 64
#define TILE_F4  32                     // float4 fragment entries per tile
#define SET_F4   (TILES_PER_SET * TILE_F4)   // 2048 float4 per set

// ---------------------------------------------------------------------------
// Prep: WMMA-ready B fragments. For each point n of each set:
//   entry[sub]    = {x, y, w, 0}  (lanes 0-15:  B rows K=0,1)
//   entry[16+sub] = {z, 1, w, 0}  (lanes 16-31: B rows K=2,3)
// w = exp2(-K2*phi_x) is the column weight (phi_x folded out of the GEMM so
// the WMMA accumulator can be the inline-0 constant).
// ---------------------------------------------------------------------------
__global__ void precompute_kernel(const float* __restrict__ C1,
                                  const float* __restrict__ C,
                                  float4* __restrict__ frag) {
    const int tid = blockIdx.x * blockDim.x + threadIdx.x;
    if (tid >= SETS * NPTS) return;
    const int s = tid >> 10;
    const int n = tid & (NPTS - 1);
    const float* src = (s < NBATCH) ? (C1 + ((size_t)s * NPTS + n) * 3)
                                    : (C + (size_t)n * 3);
    const float x = src[0], y = src[1], z = src[2];
    const float w = __builtin_amdgcn_exp2f(-K2_F * (x * x + y * y + z * z));
    const int tile = n >> 4, sub = n & 15;
    float4* dst = frag + (size_t)s * SET_F4 + (size_t)tile * TILE_F4;
    dst[sub]      = make_float4(x, y, w, 0.0f);
    dst[16 + sub] = make_float4(z, 1.0f, w, 0.0f);
}

// A fragment for a 16-row dom tile (16x4 f32 A layout):
//   lanes 0-15: {K1*dx, K1*dy}    lanes 16-31: {K1*dz, -K2*phi_m}
__device__ __forceinline__ v2f make_a(const float* __restrict__ dom,
                                      int ktile, int half, int sub) {
    const int m = ktile * 16 + sub;
    const float dx = dom[3 * m + 0];
    const float dy = dom[3 * m + 1];
    const float dz = dom[3 * m + 2];
    const float phim = dx * dx + dy * dy + dz * dz;
    v2f a;
    a.x = half ? (K1_F * dz) : (K1_F * dx);
    a.y = half ? (-K2_F * phim) : (K1_F * dy);
    return a;
}

// Hot tile: one vector load, one WMMA (C = inline 0), 8 exp2, weighted pk-FMA.
__device__ __forceinline__ void accum_tile(const float4* __restrict__ set_frag,
                                           int tile, int lane,
                                           v2f a, v8f& acc) {
    const float4 f = set_frag[tile * TILE_F4 + lane];
    v2f b;
    b.x = f.x;
    b.y = f.y;
    v8f c = {};   // inline 0 accumulator
    v8f d = __builtin_amdgcn_wmma_f32_16x16x4_f32(
        false, a, false, b, (short)0, c, false, false);
    const float w = f.z;
#pragma unroll
    for (int j = 0; j < 8; ++j)
        acc[j] += __builtin_amdgcn_exp2f(d[j]) * w;
}

// ---------------------------------------------------------------------------
// Phase 1: theta_mk over the C set. 1024 waves, one k-tile each; results to ws.
// ---------------------------------------------------------------------------
__global__ void __launch_bounds__(256)
theta_c_kernel(const float* __restrict__ dom,
               const float4* __restrict__ frag,
               float* __restrict__ theta) {          // (16384)
    const int lane  = threadIdx.x & 31;
    const int wave  = blockIdx.x * (blockDim.x >> 5) + (threadIdx.x >> 5);
    const int half  = lane >> 4;
    const int sub   = lane & 15;

    const v2f a = make_a(dom, wave, half, sub);

    const float4* fragC = frag + (size_t)NBATCH * SET_F4;
    v8f acck = {};
    for (int t = 0; t < TILES_PER_SET; ++t)
        accum_tile(fragC, t, lane, a, acck);

    // 16-lane half reduction (xor masks 1..8 stay within each half):
    // acck[j] becomes theta_mk[wave*16 + half*8 + j], replicated per half.
#pragma unroll
    for (int j = 0; j < 8; ++j) {
        float v = acck[j];
        v += __shfl_xor(v, 1);
        v += __shfl_xor(v, 2);
        v += __shfl_xor(v, 4);
        v += __shfl_xor(v, 8);
        acck[j] = v;
    }

    if (sub == 0) {
        float* dst = theta + wave * 16 + half * 8;
#pragma unroll
        for (int j = 0; j < 8; ++j)
            dst[j] = acck[j];
    }
}

// ---------------------------------------------------------------------------
// Phase 2: theta_mb folded into prod[b]. 16384 waves: wave -> (ktile, batch),
// with all 8 waves of a block sharing one batch fragment set for reuse.
// ---------------------------------------------------------------------------
__global__ void __launch_bounds__(256)
theta_b_kernel(const float* __restrict__ dom,
               const float4* __restrict__ frag,
               const float* __restrict__ theta,
               float* __restrict__ prod_acc) {       // (16)
    __shared__ float blk;
    if (threadIdx.x == 0) blk = 0.0f;
    __syncthreads();

    const int lane  = threadIdx.x & 31;
    const int wave  = blockIdx.x * (blockDim.x >> 5) + (threadIdx.x >> 5);
    const int ktile = wave & (KTILES - 1);
    const int bidx  = wave >> 10;       // same for all waves in a block
    const int half  = lane >> 4;
    const int sub   = lane & 15;

    const v2f a = make_a(dom, ktile, half, sub);

    const float4* fb = frag + (size_t)bidx * SET_F4;
    v8f accb = {};
    for (int t = 0; t < TILES_PER_SET; ++t)
        accum_tile(fb, t, lane, a, accb);

    // theta_mk for this k-tile, replicated per half (broadcast loads).
    const float* th = theta + ktile * 16 + half * 8;
    float p = 0.0f;
#pragma unroll
    for (int j = 0; j < 8; ++j)
        p += accb[j] * th[j];

    // full 32-lane reduction
    p += __shfl_xor(p, 1);
    p += __shfl_xor(p, 2);
    p += __shfl_xor(p, 4);
    p += __shfl_xor(p, 8);
    p += __shfl_xor(p, 16);

    if (lane == 0)
        atomicAdd(&blk, p);             // LDS atomic (ds_add_f32)
    __syncthreads();
    if (threadIdx.x == 0)
        atomicAdd(&prod_acc[bidx], blk); // one global atomic per block
}

__global__ void zero_acc_kernel(float* __restrict__ acc) {
    if (threadIdx.x < NBATCH) acc[threadIdx.x] = 0.0f;
}

__global__ void finalize_kernel(const float* __restrict__ acc,
                                float* __restrict__ out) {
    const int b = threadIdx.x;
    if (b < NBATCH) {
        // A * V / sqrt(n1*n2) = 8 * (1000/16384) / 1024
        const float scale = 8.0f * (1000.0f / 16384.0f) / 1024.0f;
        float dot = scale * acc[b];
        dot = fminf(fmaxf(dot, 0.0f), 1.0f);
        out[b] = 1.0f - dot;
    }
}

extern "C" void kernel_launch(void* const* d_in, const int* in_sizes, int n_in,
                              void* d_out, int out_size, void* d_ws, size_t ws_size,
                              hipStream_t stream) {
    const float* C1  = (const float*)d_in[0];   // 16*1024*3
    const float* C   = (const float*)d_in[1];   // 1024*3
    const float* dom = (const float*)d_in[2];   // 16384*3
    float* out = (float*)d_out;                 // 16

    // ws layout:
    //   [0, 256)              : 16 float prod accumulators (padded)
    //   [256, 256+64K)        : theta_mk (16384 floats)
    //   [256+64K, ...)        : SETS*SET_F4 float4 fragments (~545 KB)
    float*  acc   = (float*)d_ws;
    float*  theta = (float*)((char*)d_ws + 256);
    float4* frag  = (float4*)((char*)d_ws + 256 + NDOM * sizeof(float));

    zero_acc_kernel<<<1, 32, 0, stream>>>(acc);

    const int npts_total = SETS * NPTS;         // 17408
    precompute_kernel<<<(npts_total + 255) / 256, 256, 0, stream>>>(C1, C, frag);

    // Phase 1: 1024 waves (128 blocks x 8 waves)
    theta_c_kernel<<<KTILES / 8, 256, 0, stream>>>(dom, frag, theta);

    // Phase 2: 16384 waves (2048 blocks x 8 waves), block-uniform batch
    theta_b_kernel<<<(KTILES * NBATCH) / 8, 256, 0, stream>>>(dom, frag, theta, acc);

    finalize_kernel<<<1, 32, 0, stream>>>(acc, out);
}